// MockPolymerGCN_53455162966352
// MI455X (gfx1250) — compile-verified
//
#include <hip/hip_runtime.h>

// MockPolymerGCN segment-mean + 15->5 projection for gfx1250 (MI455X).
//
// Roofline: 240MB x-read + 16MB batch-read @ 23.3 TB/s ~= 11-12us -> pure
// bandwidth problem; compute (~60 MFLOP) is noise. Design:
//  Phase 1: stream x once. batch is sorted -> one wave per graph binary-
//           searches its node range (no atomics, deterministic). Nodes are
//           60B rows, so direct lane-strided loads scatter each VMEM op over
//           ~15 cachelines; instead we stage tiles with ASYNC global->LDS
//           b128 copies (coalesced 512B per issue, ASYNCcnt-tracked, no VGPR
//           round trip) and read rows from LDS. LDS read stride is 15 dwords
//           and gcd(15,64)=1 -> 32 lanes hit 32 distinct banks: conflict-free.
//  Phase 2: [16384,15] @ [15,5] projection with V_WMMA_F32_16X16X4_F32
//           (K padded 15->16 = four K=4 steps per 16-graph tile).

#define GRAPHS 16384
#define DIM    15
#define OUT    5
#define DPAD   16            // padded feature dim (col 15 = node count)
#define TILE_NODES 128       // nodes staged per LDS tile (7680 B)
#define LDS_FLOATS 1928      // 1920 + 16B-alignment slack (span <= 1924)

typedef __attribute__((ext_vector_type(2))) float v2f;
typedef __attribute__((ext_vector_type(8))) float v8f;

// ---------------------------------------------------------------------------
// Kernel 1: one wave (= one workgroup of 32) per graph.
// Single-wave workgroup => __syncthreads is a hardware NOP (ISA 3.1), LDS is
// private to the wave, and async/DS ordering is done with explicit waits.
// ---------------------------------------------------------------------------
__global__ void __launch_bounds__(32)
seg_mean_kernel(const float* __restrict__ x,
                const int*   __restrict__ batch,
                float*       __restrict__ meanpad,
                int n)
{
    __shared__ __attribute__((aligned(16))) float tile[LDS_FLOATS];
    const int lane = threadIdx.x;
    const int g    = blockIdx.x;

    // lower_bound(g) / lower_bound(g+1) over sorted batch[0..n)
    int lo = 0, hi = n;
    while (lo < hi) { int mid = (lo + hi) >> 1; if (batch[mid] <  g) lo = mid + 1; else hi = mid; }
    const int start = lo;
    hi = n;
    while (lo < hi) { int mid = (lo + hi) >> 1; if (batch[mid] <= g) lo = mid + 1; else hi = mid; }
    const int end = lo;

    float acc[DIM];
#pragma unroll
    for (int d = 0; d < DIM; ++d) acc[d] = 0.0f;

    const long totalF  = (long)n * DIM;
    const long maxBase = totalF - 4;                       // last valid 16B chunk
    const unsigned lds_base =
        (unsigned)(unsigned long long)(uintptr_t)&tile[0]; // low 32b = LDS offset

    const int cnt    = end - start;
    const int ntiles = (cnt + TILE_NODES - 1) / TILE_NODES;

    for (int t = 0; t < ntiles; ++t) {
        const int  s_t    = start + t * TILE_NODES;
        const int  e_t    = (s_t + TILE_NODES < end) ? (s_t + TILE_NODES) : end;
        const long flo    = (long)s_t * DIM;
        const long albase = flo & ~3L;                         // 16B-align down
        const long alend  = (((long)e_t * DIM) + 3L) & ~3L;    // 16B-align up
        const int  span   = (int)(alend - albase);             // <= 1924 floats

        // Async global->LDS staging: each issue moves 32 lanes x 16B = 512B
        // of perfectly coalesced traffic straight into LDS (ASYNCcnt).
        for (int f4 = lane; f4 * 4 < span; f4 += 32) {
            long gf = albase + (long)f4 * 4;
            if (gf > maxBase) gf = maxBase;                    // stay in-bounds
            const unsigned loff = lds_base + (unsigned)f4 * 16u;
            const unsigned long long ga =
                (unsigned long long)(uintptr_t)x + (unsigned long long)gf * 4ull;
            asm volatile("global_load_async_to_lds_b128 %0, %1, off"
                         :: "v"(loff), "v"(ga) : "memory");
        }
        asm volatile("s_wait_asynccnt 0x0" ::: "memory");
        __syncthreads();   // single-wave WG: S_NOP; keeps compiler ordering honest

        // Conflict-free LDS read-back: addr stride 15 dwords, gcd(15,64)=1.
        for (int i = s_t + lane; i < e_t; i += 32) {
            const int off = (int)((long)i * DIM - albase);
#pragma unroll
            for (int d = 0; d < DIM; ++d) acc[d] += tile[off + d];
        }
        // Retire DS reads before next tile's async writes overwrite LDS.
        asm volatile("s_wait_dscnt 0x0" ::: "memory");
        __syncthreads();
    }

    // Deterministic wave32 butterfly reduce.
#pragma unroll
    for (int d = 0; d < DIM; ++d) {
#pragma unroll
        for (int off = 16; off > 0; off >>= 1)
            acc[d] += __shfl_xor(acc[d], off, 32);
    }

    if (lane == 0) {
        const float inv = (cnt > 0) ? 1.0f / (float)cnt : 0.0f;
        float* dst = meanpad + (long)g * DPAD;
#pragma unroll
        for (int d = 0; d < DIM; ++d) dst[d] = acc[d] * inv;
        dst[DPAD - 1] = (float)cnt;       // pad slot: count (B row 15 is zero)
    }
}

// ---------------------------------------------------------------------------
// Kernel 2: one wave per 16-graph tile. D = A*B + C via
// V_WMMA_F32_16X16X4_F32, K=16 split into four K=4 chunks.
//   A 16x4 f32 (2 VGPRs): lane = M | khi=lane>>4; vgpr j holds K = 2*khi+j
//   B  4x16 f32 (2 VGPRs): N = lane&15;            vgpr j holds K = 2*khi+j
//   C/D 16x16 f32 (8 VGPRs): vgpr r -> M = r + 8*khi, N = lane&15
// B = W^T zero-padded to 16x16. Pad handled with clamped (always-valid)
// addresses + value selects -> no exec-mask branches around the WMMAs.
// ---------------------------------------------------------------------------
__global__ void __launch_bounds__(256)
proj_wmma_kernel(const float* __restrict__ meanpad,
                 const float* __restrict__ W,      // [5,15] row-major
                 const float* __restrict__ bias,   // [5]
                 float*       __restrict__ out)    // [G,5]
{
    const int wave = threadIdx.x >> 5;
    const int tile = blockIdx.x * (blockDim.x >> 5) + wave;
    const int g0   = tile * 16;
    const int lane = threadIdx.x & 31;
    const int m    = lane & 15;     // A row / B col / D col
    const int khi  = lane >> 4;

    const int wrow = (m < OUT) ? m : 0;           // clamped: load always valid

    v8f c = {};
#pragma unroll
    for (int k = 0; k < 4; ++k) {
        const int d = 4 * k + 2 * khi;            // K of vgpr 0 of this chunk
        // A chunk: aligned 8B load (row base 64B-aligned, d even).
        const v2f a = *(const v2f*)(meanpad + (long)(g0 + m) * DPAD + d);
        // B chunk: Wt[d][n] = W[n][d]; clamp addresses, select zeros.
        const float w0 = W[wrow * DIM + ((d     < DIM) ? d     : 0)];
        const float w1 = W[wrow * DIM + ((d + 1 < DIM) ? d + 1 : 0)];
        v2f b;
        b.x = (m < OUT && d     < DIM) ? w0 : 0.0f;
        b.y = (m < OUT && d + 1 < DIM) ? w1 : 0.0f;
        // (neg_a, A, neg_b, B, c_mod, C, reuse_a, reuse_b)
        c = __builtin_amdgcn_wmma_f32_16x16x4_f32(false, a, false, b,
                                                  (short)0, c, false, false);
    }

    // Columns 0..4 valid; each (graph, o) written exactly once.
    if (m < OUT) {
        const float bv = bias[m];
#pragma unroll
        for (int r = 0; r < 8; ++r) {
            const int   gg  = g0 + r + 8 * khi;
            const float nc  = meanpad[(long)gg * DPAD + (DPAD - 1)];
            out[gg * OUT + m] = (nc > 0.0f) ? (c[r] + bv) : 0.0f;
        }
    }
}

// ---------------------------------------------------------------------------
// Inputs (setup_inputs order): x[N*15] f32, batch[N] int32 (JAX canonicalizes
// int64 -> int32 without x64), W[5*15] f32, b[5] f32. Output: [16384*5] f32.
// Workspace: GRAPHS*16 floats = 1 MB padded mean matrix.
// ---------------------------------------------------------------------------
extern "C" void kernel_launch(void* const* d_in, const int* in_sizes, int n_in,
                              void* d_out, int out_size, void* d_ws, size_t ws_size,
                              hipStream_t stream) {
    const float* x       = (const float*)d_in[0];
    const int*   batch   = (const int*)  d_in[1];
    const float* W       = (const float*)d_in[2];
    const float* bias    = (const float*)d_in[3];
    float*       out     = (float*)d_out;
    float*       meanpad = (float*)d_ws;            // GRAPHS*DPAD floats

    const int n = in_sizes[0] / DIM;                // number of nodes

    // Phase 1: one wave-workgroup per graph.
    seg_mean_kernel<<<GRAPHS, 32, 0, stream>>>(x, batch, meanpad, n);

    // Phase 2: 1024 tiles of 16 graphs, 8 waves/block -> 128 blocks.
    proj_wmma_kernel<<<(GRAPHS / 16) / 8, 256, 0, stream>>>(meanpad, W, bias, out);
}